// GCN_90297392431231
// MI455X (gfx1250) — compile-verified
//
#include <hip/hip_runtime.h>
#include <hip/hip_bf16.h>

// CDNA5 / gfx1250, wave32. WMMA f32 16x16x4 for the per-vertex GEMMs,
// L2-resident float4 gathers + hardware f32 global atomics for edge scatter.

typedef float v2f __attribute__((ext_vector_type(2)));
typedef float v8f __attribute__((ext_vector_type(8)));

__device__ __forceinline__ void atomAddF32(float* p, float v) {
  // relaxed / agent scope -> HW global f32 atomic, no CAS loop
  (void)__hip_atomic_fetch_add(p, v, __ATOMIC_RELAXED, __HIP_MEMORY_SCOPE_AGENT);
}

// ---------------------------------------------------------------------------
// Phase 1: per 16-vertex tile, D(16x16) = A(16x4: xyz|0) * B(4x16: W^T|0) + bias
// ACC <- h0 (scatter accumulator), H1 <- h1 (gather source for edges).
// One wave = one tile; EXEC is all-ones at every WMMA.
// ---------------------------------------------------------------------------
__global__ void __launch_bounds__(256)
gcn_l1_gemm(const float* __restrict__ verts,
            const float* __restrict__ W0, const float* __restrict__ b0,
            const float* __restrict__ W1, const float* __restrict__ b1,
            float* __restrict__ acc, float* __restrict__ h1out, int nV) {
  const int lane = threadIdx.x & 31;
  const int wave = threadIdx.x >> 5;
  const int tile = blockIdx.x * 8 + wave;        // 16 vertices per tile
  const int row  = lane & 15;
  const bool hi  = lane >= 16;                   // K-half selector (K2,K3)
  const int vbase = tile * 16;

  // ---- A: 16x4 f32. lanes0-15: {K0,K1}={x,y}; lanes16-31: {K2,K3}={z,0}
  const int vr = min(vbase + row, nV - 1);       // select, no divergence
  const float* vp = verts + (size_t)vr * 3;
  const float x = vp[0], y = vp[1], z = vp[2];   // merges to global_load_b96
  v2f a; a.x = hi ? z : x; a.y = hi ? 0.f : y;

  // ---- B: W^T padded to 4x16. VGPR v, lane: B[v + (hi?2:0)][row]
  // W0/W1 are 16x3 row-major: W[n][k] = W[n*3+k]; B[k][n] = W[n][k]
  v2f bA, bB;
  bA.x = hi ? W0[row * 3 + 2] : W0[row * 3 + 0];
  bA.y = hi ? 0.f             : W0[row * 3 + 1];
  bB.x = hi ? W1[row * 3 + 2] : W1[row * 3 + 0];
  bB.y = hi ? 0.f             : W1[row * 3 + 1];

  // ---- C init = bias broadcast along M (bias indexed by N = row)
  const float biasA = b0[row], biasB = b1[row];
  v8f c0, c1;
#pragma unroll
  for (int i = 0; i < 8; ++i) { c0[i] = biasA; c1[i] = biasB; }

  c0 = __builtin_amdgcn_wmma_f32_16x16x4_f32(false, a, false, bA, (short)0, c0, false, false);
  c1 = __builtin_amdgcn_wmma_f32_16x16x4_f32(false, a, false, bB, (short)0, c1, false, false);

  // ---- store D: lane holds (M = r + 8*hi, N = row) for r=0..7
  // Fast path is wave-uniform (scalar branch): no per-element EXEC churn.
  const size_t base = (size_t)(vbase + (hi ? 8 : 0)) * 16 + row;
  if (vbase + 16 <= nV) {
#pragma unroll
    for (int r = 0; r < 8; ++r) {
      acc[base + (size_t)r * 16]   = c0[r];
      h1out[base + (size_t)r * 16] = c1[r];
    }
  } else {
#pragma unroll
    for (int r = 0; r < 8; ++r) {
      if (vbase + r + (hi ? 8 : 0) < nV) {
        acc[base + (size_t)r * 16]   = c0[r];
        h1out[base + (size_t)r * 16] = c1[r];
      }
    }
  }
}

// ---------------------------------------------------------------------------
// Phase 2: symmetric edge scatter of 16 features. 4 threads per edge,
// each does one float4 gather per endpoint (L2-resident: H1 = 128MB < 192MB L2)
// and 8 hardware f32 atomics into ACC.
// ---------------------------------------------------------------------------
__global__ void __launch_bounds__(256)
gcn_l1_scatter(const int* __restrict__ edges, const float* __restrict__ h1,
               float* __restrict__ acc, int nE) {
  const int t = blockIdx.x * blockDim.x + threadIdx.x;
  const int e = t >> 2;
  const int q = t & 3;
  if (e >= nE) return;
  const int src = edges[(size_t)e * 2 + 0];
  const int dst = edges[(size_t)e * 2 + 1];
  const float4 vd = *((const float4*)(h1 + (size_t)dst * 16) + q);
  const float4 vs = *((const float4*)(h1 + (size_t)src * 16) + q);
  float* as = acc + (size_t)src * 16 + q * 4;
  float* ad = acc + (size_t)dst * 16 + q * 4;
  atomAddF32(as + 0, vd.x); atomAddF32(as + 1, vd.y);
  atomAddF32(as + 2, vd.z); atomAddF32(as + 3, vd.w);
  atomAddF32(ad + 0, vs.x); atomAddF32(ad + 1, vs.y);
  atomAddF32(ad + 2, vs.z); atomAddF32(ad + 3, vs.w);
}

// ---------------------------------------------------------------------------
// Phase 3: h = relu(ACC); out = h*W0_2^T + b0_2 ; H2 = h*W1_2^T + b1_2
// K=16 via 4 chained 16x16x4 f32 WMMAs per weight matrix. Only D cols 0..2 used.
// ---------------------------------------------------------------------------
__global__ void __launch_bounds__(256)
gcn_l2_gemm(const float* __restrict__ acc,
            const float* __restrict__ W0, const float* __restrict__ b0,
            const float* __restrict__ W1, const float* __restrict__ b1,
            float* __restrict__ out, float* __restrict__ h2out, int nV) {
  const int lane = threadIdx.x & 31;
  const int wave = threadIdx.x >> 5;
  const int tile = blockIdx.x * 8 + wave;
  const int row  = lane & 15;
  const bool hi  = lane >= 16;
  const int vbase = tile * 16;

  const int vr = min(vbase + row, nV - 1);
  const float* hp = acc + (size_t)vr * 16;       // this lane's A row

  // bias along N = row; only N<3 meaningful
  const float biasA = (row < 3) ? b0[row] : 0.f;
  const float biasB = (row < 3) ? b1[row] : 0.f;
  v8f c0, c1;
#pragma unroll
  for (int i = 0; i < 8; ++i) { c0[i] = biasA; c1[i] = biasB; }

#pragma unroll
  for (int kk = 0; kk < 4; ++kk) {
    const int k0 = 4 * kk + (hi ? 2 : 0);
    // A chunk: adjacent pair -> single global_load_b64; relu applied on load
    const v2f hpair = *(const v2f*)(hp + k0);
    v2f a;
    a.x = fmaxf(hpair.x, 0.f);
    a.y = fmaxf(hpair.y, 0.f);
    // B chunks: W0_2/W1_2 are 3x16 row-major: W[n][k] = W[n*16+k]; B[k][n] = W[n][k]
    v2f bA, bB;
    bA.x = (row < 3) ? W0[row * 16 + k0 + 0] : 0.f;
    bA.y = (row < 3) ? W0[row * 16 + k0 + 1] : 0.f;
    bB.x = (row < 3) ? W1[row * 16 + k0 + 0] : 0.f;
    bB.y = (row < 3) ? W1[row * 16 + k0 + 1] : 0.f;
    c0 = __builtin_amdgcn_wmma_f32_16x16x4_f32(false, a, false, bA, (short)0, c0, false, false);
    c1 = __builtin_amdgcn_wmma_f32_16x16x4_f32(false, a, false, bB, (short)0, c1, false, false);
  }

  // store D columns 0..2 ; H2 padded to stride 4. Tile-full test is scalar.
  if (row < 3) {
    const size_t mb = (size_t)(vbase + (hi ? 8 : 0));
    if (vbase + 16 <= nV) {
#pragma unroll
      for (int r = 0; r < 8; ++r) {
        out[(mb + r) * 3 + row]   = c0[r];
        h2out[(mb + r) * 4 + row] = c1[r];
      }
    } else {
#pragma unroll
      for (int r = 0; r < 8; ++r) {
        if (vbase + r + (hi ? 8 : 0) < nV) {
          out[(mb + r) * 3 + row]   = c0[r];
          h2out[(mb + r) * 4 + row] = c1[r];
        }
      }
    }
  }
}

// ---------------------------------------------------------------------------
// Phase 4: symmetric edge scatter of 3 features into d_out.
// ---------------------------------------------------------------------------
__global__ void __launch_bounds__(256)
gcn_l2_scatter(const int* __restrict__ edges, const float* __restrict__ h2,
               float* __restrict__ out, int nE) {
  const int e = blockIdx.x * blockDim.x + threadIdx.x;
  if (e >= nE) return;
  const int src = edges[(size_t)e * 2 + 0];
  const int dst = edges[(size_t)e * 2 + 1];
  const float* hd = h2 + (size_t)dst * 4;
  const float* hs = h2 + (size_t)src * 4;
  float* os = out + (size_t)src * 3;
  float* od = out + (size_t)dst * 3;
#pragma unroll
  for (int k = 0; k < 3; ++k) {
    atomAddF32(os + k, hd[k]);
    atomAddF32(od + k, hs[k]);
  }
}

extern "C" void kernel_launch(void* const* d_in, const int* in_sizes, int n_in,
                              void* d_out, int out_size, void* d_ws, size_t ws_size,
                              hipStream_t stream) {
  const float* verts = (const float*)d_in[0];
  const int*   edges = (const int*)d_in[1];      // jnp int64 w/o x64 -> int32
  const float* W0_1  = (const float*)d_in[2];
  const float* b0_1  = (const float*)d_in[3];
  const float* W1_1  = (const float*)d_in[4];
  const float* b1_1  = (const float*)d_in[5];
  const float* W0_2  = (const float*)d_in[6];
  const float* b0_2  = (const float*)d_in[7];
  const float* W1_2  = (const float*)d_in[8];
  const float* b1_2  = (const float*)d_in[9];
  float* out = (float*)d_out;

  const int nV = in_sizes[0] / 3;                // 2,000,000
  const int nE = in_sizes[1] / 2;                // 6,000,000

  char* ws = (char*)d_ws;
  float* ACC = (float*)ws;                                        // nV*16 f32 (h0 + agg)
  float* H1  = (float*)(ws + (size_t)nV * 16 * sizeof(float));    // nV*16 f32
  float* H2  = (float*)(ws + (size_t)nV * 32 * sizeof(float));    // nV*4  f32 (padded)

  const int tiles   = (nV + 15) / 16;            // 125,000 (exact)
  const int gblocks = (tiles + 7) / 8;           // 8 waves/block, wave32

  gcn_l1_gemm<<<gblocks, 256, 0, stream>>>(verts, W0_1, b0_1, W1_1, b1_1, ACC, H1, nV);
  gcn_l1_scatter<<<(4 * nE + 255) / 256, 256, 0, stream>>>(edges, H1, ACC, nE);
  gcn_l2_gemm<<<gblocks, 256, 0, stream>>>(ACC, W0_2, b0_2, W1_2, b1_2, out, H2, nV);
  gcn_l2_scatter<<<(nE + 255) / 256, 256, 0, stream>>>(edges, H2, out, nE);
}